// MambaBackbone_26061861552219
// MI455X (gfx1250) — compile-verified
//
#include <hip/hip_runtime.h>

typedef __attribute__((ext_vector_type(2))) float v2f;
typedef __attribute__((ext_vector_type(8))) float v8f;

#define Bz   4
#define LRAW 8000
#define L1s  1600
#define C4s  128
#define Dm   512
#define Lm   800
#define NLs  4
#define Ns   16
#define DIs  1024
#define RKs  32
#define MHs  1024
#define Mrows (Bz * Lm)   // 3200

__device__ __forceinline__ float sigf(float x) { return 1.0f / (1.0f + __expf(-x)); }
__device__ __forceinline__ float siluf(float x) { return x * sigf(x); }

// Low 32 bits of a flat pointer into LDS == LDS byte offset (aperture mapping).
__device__ __forceinline__ unsigned lds_off(const void* p) {
    return (unsigned)(unsigned long long)p;
}

// ---------------------------------------------------------------------------
// WMMA GEMM:  C[M,N] = epi( A[M,K] * W[N,K]^T  (+bias) (+residual) )
// fp32 A/B via V_WMMA_F32_16X16X4_F32, fp32 accumulate (reference precision).
// Tiles staged global->LDS with GLOBAL_LOAD_ASYNC_TO_LDS_B128, double-buffered
// on ASYNCcnt; branchless software pipeline (prologue / steady loop / epilogue).
// EPI: 0=none  1=bias+softplus  2=bias+gelu(exact)  3=residual  4=residual+bias
// ---------------------------------------------------------------------------
#define GTM 64
#define GTN 64
#define GKB 32

template <int EPI>
__global__ __launch_bounds__(256) void gemm_wmma(
    const float* __restrict__ A, int lda,
    const float* __restrict__ W, int ldw,
    const float* __restrict__ bias,
    const float* __restrict__ R, int ldr,
    float* __restrict__ C, int ldc,
    int M, int N, int K)
{
    __shared__ float As[2][GTM][GKB + 4];
    __shared__ float Ws[2][GTN][GKB + 4];

    const int tid  = threadIdx.x;
    const int m0   = blockIdx.x * GTM;
    const int n0   = blockIdx.y * GTN;
    const int wave = tid >> 5;
    const int lane = tid & 31;
    const int wm   = (wave >> 1) * 16;   // 0,16,32,48
    const int wn   = (wave & 1) * 32;    // 0 or 32 -> two 16-wide tiles
    const int hi   = lane >> 4;          // half-select (K pair / M+8)
    const int fr   = lane & 15;          // row (A) / col (B,C)

    // staging coordinates: 512 float4 per tile across 256 threads (2 each)
    const int sr0 = tid >> 3;                 // rows 0..31
    const int sc0 = (tid & 7) << 2;           // col 0..28 step 4
    const int sr1 = sr0 + 32;                 // rows 32..63

    v8f c0 = {};
    v8f c1 = {};

    const int nslab = K / GKB;

    // issue async global->LDS copies for one K-slab into buffer `buf`
    auto issue = [&](int buf, int k0) {
        unsigned la0 = lds_off(&As[buf][sr0][sc0]);
        unsigned la1 = lds_off(&As[buf][sr1][sc0]);
        unsigned lw0 = lds_off(&Ws[buf][sr0][sc0]);
        unsigned lw1 = lds_off(&Ws[buf][sr1][sc0]);
        const float* ga0 = &A[(size_t)(m0 + sr0) * lda + k0 + sc0];
        const float* ga1 = &A[(size_t)(m0 + sr1) * lda + k0 + sc0];
        const float* gw0 = &W[(size_t)(n0 + sr0) * ldw + k0 + sc0];
        const float* gw1 = &W[(size_t)(n0 + sr1) * ldw + k0 + sc0];
        asm volatile("global_load_async_to_lds_b128 %0, %1, off"
                     :: "v"(la0), "v"(ga0) : "memory");
        asm volatile("global_load_async_to_lds_b128 %0, %1, off"
                     :: "v"(lw0), "v"(gw0) : "memory");
        asm volatile("global_load_async_to_lds_b128 %0, %1, off"
                     :: "v"(la1), "v"(ga1) : "memory");
        asm volatile("global_load_async_to_lds_b128 %0, %1, off"
                     :: "v"(lw1), "v"(gw1) : "memory");
    };

    // 16 WMMA ops on one resident K-slab
    auto compute = [&](int buf) {
#pragma unroll
        for (int kk = 0; kk < GKB; kk += 4) {
            v2f a  = *(const v2f*)&As[buf][wm + fr][kk + 2 * hi];
            v2f b0 = *(const v2f*)&Ws[buf][wn + fr][kk + 2 * hi];
            v2f b1 = *(const v2f*)&Ws[buf][wn + 16 + fr][kk + 2 * hi];
            c0 = __builtin_amdgcn_wmma_f32_16x16x4_f32(false, a, false, b0, (short)0, c0, false, false);
            c1 = __builtin_amdgcn_wmma_f32_16x16x4_f32(false, a, false, b1, (short)0, c1, false, false);
        }
    };

    // prologue
    issue(0, 0);
    // steady state: fetch slab s+1 while matrix pipe runs slab s
    int s = 0;
    for (; s < nslab - 1; ++s) {
        const int buf = s & 1;
        issue(buf ^ 1, (s + 1) * GKB);
        // in-order async completion: <=4 outstanding => slab s has landed
        asm volatile("s_wait_asynccnt 0x4" ::: "memory");
        __syncthreads();
        compute(buf);
        __syncthreads();   // all waves done reading buf -> safe to overwrite
    }
    // epilogue: last slab
    asm volatile("s_wait_asynccnt 0x0" ::: "memory");
    __syncthreads();
    compute(s & 1);

    const int colA = n0 + wn + fr;
    const int colB = colA + 16;
    float bA = 0.0f, bB = 0.0f;
    if (EPI == 1 || EPI == 2 || EPI == 4) { bA = bias[colA]; bB = bias[colB]; }

#pragma unroll
    for (int r = 0; r < 8; ++r) {
        int row  = m0 + wm + hi * 8 + r;
        float v0 = c0[r];
        float v1 = c1[r];
        if (EPI == 1) {                       // bias + softplus
            v0 += bA; v1 += bB;
            v0 = (v0 > 20.0f) ? v0 : log1pf(__expf(v0));
            v1 = (v1 > 20.0f) ? v1 : log1pf(__expf(v1));
        } else if (EPI == 2) {                // bias + exact gelu
            v0 += bA; v1 += bB;
            v0 = 0.5f * v0 * (1.0f + erff(v0 * 0.70710678118654752f));
            v1 = 0.5f * v1 * (1.0f + erff(v1 * 0.70710678118654752f));
        } else if (EPI == 3) {                // residual
            v0 += R[(size_t)row * ldr + colA];
            v1 += R[(size_t)row * ldr + colB];
        } else if (EPI == 4) {                // residual + bias
            v0 += R[(size_t)row * ldr + colA] + bA;
            v1 += R[(size_t)row * ldr + colB] + bB;
        }
        C[(size_t)row * ldc + colA] = v0;
        C[(size_t)row * ldc + colB] = v1;
    }
}

// ---------------------------------------------------------------------------
// Conv stem 1: 1->128 ch, k=10, s=5, pad=3, + BN(eval) + SiLU. out [B,128,1600]
// ---------------------------------------------------------------------------
__global__ __launch_bounds__(256) void conv_stem1(
    const float* __restrict__ x, const float* __restrict__ w, const float* __restrict__ bc,
    const float* __restrict__ g, const float* __restrict__ bb,
    const float* __restrict__ m, const float* __restrict__ v,
    float* __restrict__ o)
{
    int idx = blockIdx.x * 256 + threadIdx.x;      // (b, c, t), t fastest
    int t = idx % L1s;
    int c = (idx / L1s) % C4s;
    int b = idx / (L1s * C4s);
    float acc = 0.0f;
#pragma unroll
    for (int k = 0; k < 10; ++k) {
        int p = 5 * t + k - 3;
        if (p >= 0 && p < LRAW) acc += x[(size_t)b * LRAW + p] * w[c * 10 + k];
    }
    acc += bc[c];
    acc = (acc - m[c]) * rsqrtf(v[c] + 1e-5f) * g[c] + bb[c];
    o[idx] = siluf(acc);
}

// ---------------------------------------------------------------------------
// Conv stem 2: 128->512 ch, k=3, s=2, pad=1, + BN, store transposed [B,L,D]
// ---------------------------------------------------------------------------
__global__ __launch_bounds__(256) void conv_stem2(
    const float* __restrict__ h1, const float* __restrict__ w, const float* __restrict__ bc,
    const float* __restrict__ g, const float* __restrict__ bb,
    const float* __restrict__ m, const float* __restrict__ v,
    float* __restrict__ o)
{
    int idx = blockIdx.x * 256 + threadIdx.x;      // (b, l, d), d fastest
    int d = idx % Dm;
    int l = (idx / Dm) % Lm;
    int b = idx / (Dm * Lm);
    float acc = 0.0f;
    for (int c = 0; c < C4s; ++c) {
        const float* hr = h1 + ((size_t)b * C4s + c) * L1s;
        const float* wr = w + ((size_t)d * C4s + c) * 3;
#pragma unroll
        for (int k = 0; k < 3; ++k) {
            int p = 2 * l + k - 1;
            if (p >= 0 && p < L1s) acc += hr[p] * wr[k];
        }
    }
    acc += bc[d];
    acc = (acc - m[d]) * rsqrtf(v[d] + 1e-5f) * g[d] + bb[d];
    o[((size_t)b * Lm + l) * Dm + d] = acc;
}

// ---------------------------------------------------------------------------
// LayerNorm over D=512; one 256-thread block per row
// ---------------------------------------------------------------------------
__global__ __launch_bounds__(256) void ln512(
    const float* __restrict__ x, const float* __restrict__ g,
    const float* __restrict__ b, float* __restrict__ o)
{
    __shared__ float red[256];
    int row = blockIdx.x;
    int tid = threadIdx.x;
    const float* xr = x + (size_t)row * Dm;
    float v0 = xr[tid], v1 = xr[tid + 256];

    red[tid] = v0 + v1;
    __syncthreads();
    for (int s = 128; s > 0; s >>= 1) { if (tid < s) red[tid] += red[tid + s]; __syncthreads(); }
    float mu = red[0] * (1.0f / 512.0f);
    __syncthreads();

    float d0 = v0 - mu, d1 = v1 - mu;
    red[tid] = d0 * d0 + d1 * d1;
    __syncthreads();
    for (int s = 128; s > 0; s >>= 1) { if (tid < s) red[tid] += red[tid + s]; __syncthreads(); }
    float rstd = rsqrtf(red[0] * (1.0f / 512.0f) + 1e-5f);

    o[(size_t)row * Dm + tid]       = d0 * rstd * g[tid] + b[tid];
    o[(size_t)row * Dm + tid + 256] = d1 * rstd * g[tid + 256] + b[tid + 256];
}

// ---------------------------------------------------------------------------
// Depthwise causal conv (DC=4) + bias + SiLU on xm (first half of xz)
// ---------------------------------------------------------------------------
__global__ __launch_bounds__(256) void dwconv_silu(
    const float* __restrict__ xz, const float* __restrict__ cw,
    const float* __restrict__ cb, float* __restrict__ xc)
{
    int idx = blockIdx.x * 256 + threadIdx.x;      // (b, l, d), d fastest
    int d = idx % DIs;
    int l = (idx / DIs) % Lm;
    int b = idx / (DIs * Lm);
    float acc = 0.0f;
#pragma unroll
    for (int j = 0; j < 4; ++j) {
        int lp = l + j - 3;
        if (lp >= 0) acc += xz[((size_t)(b * Lm + lp)) * (2 * DIs) + d] * cw[d * 4 + j];
    }
    acc += cb[d];
    xc[((size_t)(b * Lm + l)) * DIs + d] = siluf(acc);
}

// ---------------------------------------------------------------------------
// Selective scan, fused with +x*D and silu(z) gate.
// grid (DI/256, B); thread owns (b,d): state h[16] in registers.
// ---------------------------------------------------------------------------
__global__ __launch_bounds__(256) void scan_fused(
    const float* __restrict__ dbl, const float* __restrict__ dtb,
    const float* __restrict__ xc,  const float* __restrict__ xz,
    const float* __restrict__ A_log, const float* __restrict__ Dp,
    float* __restrict__ y)
{
    __shared__ float bcS[32];                    // B_t[0..15], C_t[0..15]
    int b = blockIdx.y;
    int d = blockIdx.x * 256 + threadIdx.x;
    int tid = threadIdx.x;

    float A[Ns];
#pragma unroll
    for (int n = 0; n < Ns; ++n) A[n] = -__expf(A_log[(size_t)d * Ns + n]);
    float Dpd = Dp[d];
    float h[Ns];
#pragma unroll
    for (int n = 0; n < Ns; ++n) h[n] = 0.0f;

    for (int l = 0; l < Lm; ++l) {
        size_t base = (size_t)b * Lm + l;
        if (tid < 32) bcS[tid] = dbl[base * 64 + RKs + tid];
        __syncthreads();

        float dt  = dtb[base * DIs + d];
        float xv  = xc[base * DIs + d];
        float dtx = dt * xv;
        float acc = 0.0f;
#pragma unroll
        for (int n = 0; n < Ns; ++n) {
            h[n] = h[n] * __expf(dt * A[n]) + dtx * bcS[n];
            acc += h[n] * bcS[16 + n];
        }
        float z = xz[base * (2 * DIs) + DIs + d];
        y[base * DIs + d] = (acc + xv * Dpd) * siluf(z);
        __syncthreads();
    }
}

// ---------------------------------------------------------------------------
// Mean over L of the final-LN output -> out [B, D]
// ---------------------------------------------------------------------------
__global__ __launch_bounds__(256) void mean_over_l(
    const float* __restrict__ t, float* __restrict__ out)
{
    int idx = blockIdx.x * 256 + threadIdx.x;    // (b, d)
    int d = idx % Dm;
    int b = idx / Dm;
    float s = 0.0f;
    for (int l = 0; l < Lm; ++l) s += t[((size_t)b * Lm + l) * Dm + d];
    out[idx] = s * (1.0f / (float)Lm);
}

// ---------------------------------------------------------------------------
extern "C" void kernel_launch(void* const* d_in, const int* in_sizes, int n_in,
                              void* d_out, int out_size, void* d_ws, size_t ws_size,
                              hipStream_t stream)
{
    (void)in_sizes; (void)n_in; (void)out_size; (void)ws_size;

    const float* x    = (const float*)d_in[0];
    const float* c1w  = (const float*)d_in[1];
    const float* c1b  = (const float*)d_in[2];
    const float* bn1g = (const float*)d_in[3];
    const float* bn1b = (const float*)d_in[4];
    const float* bn1m = (const float*)d_in[5];
    const float* bn1v = (const float*)d_in[6];
    const float* c2w  = (const float*)d_in[7];
    const float* c2b  = (const float*)d_in[8];
    const float* bn2g = (const float*)d_in[9];
    const float* bn2b = (const float*)d_in[10];
    const float* bn2m = (const float*)d_in[11];
    const float* bn2v = (const float*)d_in[12];
    const float* ln1g = (const float*)d_in[13];
    const float* ln1b = (const float*)d_in[14];
    const float* inw  = (const float*)d_in[15];
    const float* cw   = (const float*)d_in[16];
    const float* cb   = (const float*)d_in[17];
    const float* xpw  = (const float*)d_in[18];
    const float* dtw  = (const float*)d_in[19];
    const float* dtbi = (const float*)d_in[20];
    const float* Alog = (const float*)d_in[21];
    const float* Dp   = (const float*)d_in[22];
    const float* outw = (const float*)d_in[23];
    const float* ln2g = (const float*)d_in[24];
    const float* ln2b = (const float*)d_in[25];
    const float* m1w  = (const float*)d_in[26];
    const float* m1b  = (const float*)d_in[27];
    const float* m2w  = (const float*)d_in[28];
    const float* m2b  = (const float*)d_in[29];
    const float* fng  = (const float*)d_in[30];
    const float* fnb  = (const float*)d_in[31];

    float* ws   = (float*)d_ws;
    float* h    = ws;                                  // [3200, 512]
    float* tA   = h    + (size_t)Mrows * Dm;           // [3200, 512]  (ln scratch)
    float* xzb  = tA   + (size_t)Mrows * Dm;           // [3200, 2048]
    float* xcb  = xzb  + (size_t)Mrows * 2 * DIs;      // [3200, 1024]
    float* dblb = xcb  + (size_t)Mrows * DIs;          // [3200, 64]
    float* dtb  = dblb + (size_t)Mrows * 64;           // [3200, 1024]
    float* yb   = dtb  + (size_t)Mrows * DIs;          // [3200, 1024]
    float* h1   = xzb;                                 // conv-stem scratch (reuse)
    float* mm   = xzb;                                 // MLP hidden (reuse)

    conv_stem1<<<(Bz * C4s * L1s) / 256, 256, 0, stream>>>(x, c1w, c1b, bn1g, bn1b, bn1m, bn1v, h1);
    conv_stem2<<<(Bz * Lm * Dm) / 256, 256, 0, stream>>>(h1, c2w, c2b, bn2g, bn2b, bn2m, bn2v, h);

    for (int i = 0; i < NLs; ++i) {
        const float* ln1g_i = ln1g + (size_t)i * Dm;
        const float* ln1b_i = ln1b + (size_t)i * Dm;
        const float* inw_i  = inw  + (size_t)i * 2 * DIs * Dm;
        const float* cw_i   = cw   + (size_t)i * DIs * 4;
        const float* cb_i   = cb   + (size_t)i * DIs;
        const float* xpw_i  = xpw  + (size_t)i * 64 * DIs;
        const float* dtw_i  = dtw  + (size_t)i * DIs * RKs;
        const float* dtb_i  = dtbi + (size_t)i * DIs;
        const float* Alog_i = Alog + (size_t)i * DIs * Ns;
        const float* Dp_i   = Dp   + (size_t)i * DIs;
        const float* outw_i = outw + (size_t)i * Dm * DIs;
        const float* ln2g_i = ln2g + (size_t)i * Dm;
        const float* ln2b_i = ln2b + (size_t)i * Dm;
        const float* m1w_i  = m1w  + (size_t)i * MHs * Dm;
        const float* m1b_i  = m1b  + (size_t)i * MHs;
        const float* m2w_i  = m2w  + (size_t)i * Dm * MHs;
        const float* m2b_i  = m2b  + (size_t)i * Dm;

        // u = LN(h)
        ln512<<<Mrows, 256, 0, stream>>>(h, ln1g_i, ln1b_i, tA);
        // xz = u @ in_w^T   [3200, 2048]
        gemm_wmma<0><<<dim3(Mrows / 64, (2 * DIs) / 64), 256, 0, stream>>>(
            tA, Dm, inw_i, Dm, nullptr, nullptr, 0, xzb, 2 * DIs, Mrows, 2 * DIs, Dm);
        // xc = silu(dwconv(xm) + cb)
        dwconv_silu<<<(Bz * Lm * DIs) / 256, 256, 0, stream>>>(xzb, cw_i, cb_i, xcb);
        // dbl = xc @ xp_w^T  [3200, 64]
        gemm_wmma<0><<<dim3(Mrows / 64, 1), 256, 0, stream>>>(
            xcb, DIs, xpw_i, DIs, nullptr, nullptr, 0, dblb, 64, Mrows, 64, DIs);
        // dt = softplus(dtr @ dt_w^T + dt_b)  [3200, 1024]
        gemm_wmma<1><<<dim3(Mrows / 64, DIs / 64), 256, 0, stream>>>(
            dblb, 64, dtw_i, RKs, dtb_i, nullptr, 0, dtb, DIs, Mrows, DIs, RKs);
        // y = scan(...) fused with +x*D and silu(z) gate
        scan_fused<<<dim3(DIs / 256, Bz), 256, 0, stream>>>(
            dblb, dtb, xcb, xzb, Alog_i, Dp_i, yb);
        // h = h + y @ out_w^T
        gemm_wmma<3><<<dim3(Mrows / 64, Dm / 64), 256, 0, stream>>>(
            yb, DIs, outw_i, DIs, nullptr, h, Dm, h, Dm, Mrows, Dm, DIs);
        // m = LN(h); mm = gelu(m @ m1_w^T + m1_b); h = h + mm @ m2_w^T + m2_b
        ln512<<<Mrows, 256, 0, stream>>>(h, ln2g_i, ln2b_i, tA);
        gemm_wmma<2><<<dim3(Mrows / 64, MHs / 64), 256, 0, stream>>>(
            tA, Dm, m1w_i, Dm, m1b_i, nullptr, 0, mm, MHs, Mrows, MHs, Dm);
        gemm_wmma<4><<<dim3(Mrows / 64, Dm / 64), 256, 0, stream>>>(
            mm, MHs, m2w_i, MHs, m2b_i, h, Dm, h, Dm, Mrows, Dm, MHs);
    }

    ln512<<<Mrows, 256, 0, stream>>>(h, fng, fnb, tA);
    mean_over_l<<<(Bz * Dm) / 256, 256, 0, stream>>>(tA, (float*)d_out);
}